// GaussianSuperposition_28733331210408
// MI455X (gfx1250) — compile-verified
//
#include <hip/hip_runtime.h>
#include <hip/hip_bf16.h>

typedef __attribute__((ext_vector_type(2))) float v2f;
typedef __attribute__((ext_vector_type(8))) float v8f;

#define N_    64
#define C_    4
#define D_    4096
#define G_    32          // gaussians per (n,c) slab = LPC*KPL
#define P_    384

#define WAVES_PER_BLOCK 8
#define THREADS (WAVES_PER_BLOCK * 32)
#define DCHUNKS 4
#define D_PER_BLOCK (D_ / DCHUNKS)                    // 1024
#define D_PER_WAVE  (D_PER_BLOCK / WAVES_PER_BLOCK)   // 128
#define PAIRS       (D_PER_WAVE / 32)                 // 4 (two 16-d tiles per pair)

#define SWAPX16 0x401f   // ds_swizzle group-of-32: xor=0x10, or=0, and=0x1f

__device__ __forceinline__ float swap_halves(float v) {
    return __int_as_float(__builtin_amdgcn_ds_swizzle(__float_as_int(v), SWAPX16));
}

__global__ __launch_bounds__(THREADS)
void gauss_superpos_wmma(const float* __restrict__ x,
                         const float* __restrict__ net,
                         const int*   __restrict__ amp_idx,
                         const int*   __restrict__ mu_idx,
                         const int*   __restrict__ sigma_idx,
                         float*       __restrict__ out)
{
    __shared__ float sA2[G_], sA1[G_], sA0[G_], sAmp[G_];

    const int nc = blockIdx.x;          // 0..255 -> (n, c)
    const int n  = nc / C_;
    const int c  = nc % C_;
    const int tid = threadIdx.x;

    // ---- per-(n,c) coefficient build: exponent(base2) = A2*x^2 + A1*x + A0 ----
    if (tid < G_) {
        const int g  = tid;
        const int ia = amp_idx  [c * G_ + g];
        const int im = mu_idx   [c * G_ + g];
        const int is = sigma_idx[c * G_ + g];
        const float amp = net[n * P_ + ia];
        const float mu  = net[n * P_ + im];
        const float sg  = net[n * P_ + is];
        const float iv  = 1.0f / (sg * sg);
        const float LOG2E = 1.4426950408889634f;
        const float h = -0.5f * iv * LOG2E;
        sA2[g]  = h;
        sA1[g]  = mu * iv * LOG2E;
        sA0[g]  = h * mu * mu;
        sAmp[g] = amp;
    }
    __syncthreads();

    const int lane = tid & 31;
    const int half = lane >> 4;      // 0: lanes 0-15, 1: lanes 16-31
    const int gl   = lane & 15;
    const int wave = tid >> 5;

    // A operands: 16x4 f32 (M=g rows, K cols [A2, A1, A0, 0]).
    // Lanes 0-15 hold K0(v0)/K1(v1); lanes 16-31 hold K2(v0)/K3(v1).
    v2f At0, At1;
    At0.x = half ? sA0[gl]      : sA2[gl];
    At0.y = half ? 0.0f         : sA1[gl];
    At1.x = half ? sA0[16 + gl] : sA2[16 + gl];
    At1.y = half ? 0.0f         : sA1[16 + gl];

    // amp per D-tile row: rows g = r + 8*half (tile of g 0..15), +16 (g 16..31)
    float amp0[8], amp1[8];
#pragma unroll
    for (int r = 0; r < 8; ++r) {
        amp0[r] = sAmp[r + 8 * half];
        amp1[r] = sAmp[16 + r + 8 * half];
    }

    const float* __restrict__ xrow = x   + (size_t)nc * D_;
    float*       __restrict__ orow = out + (size_t)nc * D_;
    const int dbase = blockIdx.y * D_PER_BLOCK + wave * D_PER_WAVE;

    // Hoist all loads: 4 full-width coalesced b32 loads cover the 128-d span.
    float xv[PAIRS];
#pragma unroll
    for (int p = 0; p < PAIRS; ++p)
        xv[p] = xrow[dbase + p * 32 + lane];

#pragma unroll
    for (int p = 0; p < PAIRS; ++p) {
        // tile0: d = dbase+p*32+gl (x in lanes 0-15); tile1: +16 (x in lanes 16-31)
        const float xlo = xv[p];
        const float xhi = swap_halves(xv[p]);   // upper half's x -> lower lanes

        // B operands: 4x16 f32 (K rows x N=d): rows [x^2, x, 1, 0]
        v2f B0, B1;
        B0.x = half ? 1.0f : xlo * xlo;  B0.y = half ? 0.0f : xlo;
        B1.x = half ? 1.0f : xhi * xhi;  B1.y = half ? 0.0f : xhi;
        v8f cz = {0.f, 0.f, 0.f, 0.f, 0.f, 0.f, 0.f, 0.f};

        v8f e00 = __builtin_amdgcn_wmma_f32_16x16x4_f32(
                      false, At0, false, B0, (short)0, cz, false, false);
        v8f e01 = __builtin_amdgcn_wmma_f32_16x16x4_f32(
                      false, At1, false, B0, (short)0, cz, false, false);
        v8f e10 = __builtin_amdgcn_wmma_f32_16x16x4_f32(
                      false, At0, false, B1, (short)0, cz, false, false);
        v8f e11 = __builtin_amdgcn_wmma_f32_16x16x4_f32(
                      false, At1, false, B1, (short)0, cz, false, false);

        // amp-weighted exp2 reductions; two accumulators per tile for ILP
        float s0a = 0.f, s0b = 0.f, s1a = 0.f, s1b = 0.f;
#pragma unroll
        for (int r = 0; r < 8; ++r) {
            s0a = fmaf(amp0[r], __builtin_amdgcn_exp2f(e00[r]), s0a);
            s0b = fmaf(amp1[r], __builtin_amdgcn_exp2f(e01[r]), s0b);
        }
#pragma unroll
        for (int r = 0; r < 8; ++r) {
            s1a = fmaf(amp0[r], __builtin_amdgcn_exp2f(e10[r]), s1a);
            s1b = fmaf(amp1[r], __builtin_amdgcn_exp2f(e11[r]), s1b);
        }
        const float s0 = s0a + s0b;           // tile0 partial (this lane-half's 16 g)
        const float s1 = s1a + s1b;           // tile1 partial
        const float tot0 = s0 + swap_halves(s0);   // full 32-g sum, tile0 column gl
        const float tot1 = s1 + swap_halves(s1);   // full 32-g sum, tile1 column gl

        // Full-width, non-divergent store: lanes<16 -> tile0, lanes>=16 -> tile1
        orow[dbase + p * 32 + lane] = half ? tot1 : tot0;
    }
}

extern "C" void kernel_launch(void* const* d_in, const int* in_sizes, int n_in,
                              void* d_out, int out_size, void* d_ws, size_t ws_size,
                              hipStream_t stream) {
    const float* x   = (const float*)d_in[0];
    const float* net = (const float*)d_in[1];
    const int*   ai  = (const int*)d_in[2];
    const int*   mi  = (const int*)d_in[3];
    const int*   si  = (const int*)d_in[4];
    float* out = (float*)d_out;

    dim3 grid(N_ * C_, DCHUNKS);
    gauss_superpos_wmma<<<grid, THREADS, 0, stream>>>(x, net, ai, mi, si, out);
    (void)in_sizes; (void)n_in; (void)out_size; (void)d_ws; (void)ws_size;
}